// PointNetSampler_11433202942130
// MI455X (gfx1250) — compile-verified
//
#include <hip/hip_runtime.h>

typedef __attribute__((ext_vector_type(2))) float v2f;
typedef __attribute__((ext_vector_type(8))) float v8f;

// Problem dims (from reference setup_inputs)
constexpr int Bdim = 4, Npts = 16384, Mctr = 1024, Dxyz = 3, Cfeat = 64;
constexpr int Cop = 64, Cout = 128, Kn = 32;
constexpr int CIN = 68;          // 3 + 64 = 67, padded to 68 (K % 4 == 0)
constexpr float R2 = 0.25f;      // RADIUS^2
constexpr int WAVES2 = 4;        // waves per block in fused kernel

// ---------------------------------------------------------------------------
// Kernel 1: ball query. One wave32 per (b,m) distance row. Scan in order,
// compact the first K passing indices via ballot + prefix popcount, pad with
// N. Early exit => reads ~160 of 16384 elements per row on average.
// ---------------------------------------------------------------------------
__global__ __launch_bounds__(256) void ball_query_kernel(
    const float* __restrict__ dist, int* __restrict__ ids) {
  const int lane = threadIdx.x & 31;
  const int row = blockIdx.x * (blockDim.x >> 5) + (threadIdx.x >> 5);
  if (row >= Bdim * Mctr) return;

  const float* d = dist + (size_t)row * Npts;
  int* out = ids + (size_t)row * Kn;

  int count = 0;
  for (int n0 = 0; n0 < Npts && count < Kn; n0 += 32) {
    if (n0 + 128 < Npts) __builtin_prefetch(&d[n0 + 128 + lane], 0, 0);
    const bool pass = d[n0 + lane] < R2;
    const unsigned bal = (unsigned)__ballot(pass);           // wave32: low 32 bits
    const int slot = count + __popc(bal & ((1u << lane) - 1u));
    if (pass && slot < Kn) out[slot] = n0 + lane;
    count += __popc(bal);
  }
  if (count > Kn) count = Kn;
  if (lane >= count) out[lane] = Npts;                       // pad with N
}

// ---------------------------------------------------------------------------
// Kernel 2: fused gather + neighbor MLP (WMMA f32 16x16x4) + masked max-pool
// + per-center 64x128 linear + ReLU. One center per wave.
// ---------------------------------------------------------------------------
__global__ __launch_bounds__(32 * WAVES2) void pointnet_fused_kernel(
    const float* __restrict__ positions, const float* __restrict__ features,
    const float* __restrict__ centers, const int* __restrict__ ids,
    const float* __restrict__ W_op, const float* __restrict__ b_op,
    const float* __restrict__ W_agg, const float* __restrict__ b_agg,
    float* __restrict__ out) {
  // Staged weights, TRANSPOSED: sWt[c][r] with input rows reordered so that
  // r 0..63 = feature rows (W_op rows 3..66), r 64..66 = rel-pos rows
  // (W_op rows 0..2), r 67 = zero pad. Transposed storage makes each WMMA
  // B-fragment {W[r][c], W[r+1][c]} a single contiguous 8B ds_load_b64.
  __shared__ float sWt[Cop][CIN];
  __shared__ float sBop[Cop];
  __shared__ float sA[WAVES2][Kn][CIN];
  __shared__ float sPool[WAVES2][Cop];

  const int tid = threadIdx.x;
  for (int i = tid; i < CIN * Cop; i += blockDim.x) {
    const int r = i / Cop, c = i - r * Cop;
    float v = 0.0f;
    if (r < Cfeat)            v = W_op[(size_t)(3 + r) * Cop + c];
    else if (r < Cfeat + 3)   v = W_op[(size_t)(r - Cfeat) * Cop + c];
    sWt[c][r] = v;
  }
  if (tid < Cop) sBop[tid] = b_op[tid];

  const int wave = tid >> 5, lane = tid & 31;
  const int cid = blockIdx.x * WAVES2 + wave;   // grid sized so cid < B*M
  const int b = cid / Mctr;
  const int m = cid - b * Mctr;

  // Neighbor id for this lane's slot (lane == slot k).
  const int id = ids[(size_t)cid * Kn + lane];
  const bool valid = id < Npts;
  const unsigned vmask = (unsigned)__ballot(valid);
  const int safe = valid ? id : 0;

  // Build A row `lane`: [feat(64) | rel(3) | 0]  (16B-aligned float4 stores)
  float* arow = &sA[wave][lane][0];
  const float* fp = features + ((size_t)b * Npts + safe) * Cfeat;
#pragma unroll
  for (int j = 0; j < Cfeat; j += 4)
    *reinterpret_cast<float4*>(arow + j) = *reinterpret_cast<const float4*>(fp + j);
  const float* pp = positions + ((size_t)b * Npts + safe) * Dxyz;
  const float* cc = centers + ((size_t)b * Mctr + m) * Dxyz;
  arow[Cfeat + 0] = pp[0] - cc[0];
  arow[Cfeat + 1] = pp[1] - cc[1];
  arow[Cfeat + 2] = pp[2] - cc[2];
  arow[Cfeat + 3] = 0.0f;

  __syncthreads();

  // f(32x64) = A(32x68) @ W(68x64) via V_WMMA_F32_16X16X4_F32
  // A 16x4 f32 layout: lanes 0-15 hold K=k0,k0+1; lanes 16-31 hold K=k0+2,k0+3.
  // B 4x16 layout mirrors it: VGPR pair {K=k0+2hi, K=k0+2hi+1} at column lm.
  const int hi = lane >> 4, lm = lane & 15;
  v8f zero = {};
  v8f acc[2][4];
#pragma unroll
  for (int mt = 0; mt < 2; ++mt)
#pragma unroll
    for (int nt = 0; nt < 4; ++nt) acc[mt][nt] = zero;

#pragma unroll
  for (int k0 = 0; k0 < CIN; k0 += 4) {
    const v2f a0 = *reinterpret_cast<const v2f*>(&sA[wave][lm][k0 + 2 * hi]);
    const v2f a1 = *reinterpret_cast<const v2f*>(&sA[wave][16 + lm][k0 + 2 * hi]);
#pragma unroll
    for (int nt = 0; nt < 4; ++nt) {
      const v2f bf =
          *reinterpret_cast<const v2f*>(&sWt[nt * 16 + lm][k0 + 2 * hi]);
      acc[0][nt] = __builtin_amdgcn_wmma_f32_16x16x4_f32(
          false, a0, false, bf, (short)0, acc[0][nt], false, false);
      acc[1][nt] = __builtin_amdgcn_wmma_f32_16x16x4_f32(
          false, a1, false, bf, (short)0, acc[1][nt], false, false);
    }
  }

  // Bias + zero invalid slots + max-pool over the 32 neighbor slots.
  // C/D layout: VGPR v, lanes 0-15 -> row v, lanes 16-31 -> row 8+v; col = lm.
#pragma unroll
  for (int nt = 0; nt < 4; ++nt) {
    const float bia = sBop[nt * 16 + lm];
    float mx = -3.402823466e+38f;
#pragma unroll
    for (int mt = 0; mt < 2; ++mt)
#pragma unroll
      for (int v = 0; v < 8; ++v) {
        const int slotRow = mt * 16 + hi * 8 + v;
        const float fv =
            ((vmask >> slotRow) & 1u) ? (acc[mt][nt][v] + bia) : 0.0f;
        mx = fmaxf(mx, fv);
      }
    mx = fmaxf(mx, __shfl_xor(mx, 16));   // combine rows 0-7 with 8-15 halves
    if (hi == 0) sPool[wave][nt * 16 + lm] = mx;
  }
  __syncthreads();

  // out = relu(pooled @ W_agg + b_agg); each lane computes 4 output channels.
  const float4 ba = *reinterpret_cast<const float4*>(b_agg + 4 * lane);
  float o0 = ba.x, o1 = ba.y, o2 = ba.z, o3 = ba.w;
#pragma unroll 4
  for (int i = 0; i < Cop; ++i) {
    const float p = sPool[wave][i];
    const float4 w =
        *reinterpret_cast<const float4*>(W_agg + (size_t)i * Cout + 4 * lane);
    o0 = fmaf(p, w.x, o0);
    o1 = fmaf(p, w.y, o1);
    o2 = fmaf(p, w.z, o2);
    o3 = fmaf(p, w.w, o3);
  }
  float4 r;
  r.x = fmaxf(o0, 0.0f);
  r.y = fmaxf(o1, 0.0f);
  r.z = fmaxf(o2, 0.0f);
  r.w = fmaxf(o3, 0.0f);
  *reinterpret_cast<float4*>(out + (size_t)cid * Cout + 4 * lane) = r;
}

// ---------------------------------------------------------------------------
extern "C" void kernel_launch(void* const* d_in, const int* in_sizes, int n_in,
                              void* d_out, int out_size, void* d_ws,
                              size_t ws_size, hipStream_t stream) {
  const float* positions = (const float*)d_in[0];
  const float* features  = (const float*)d_in[1];
  const float* centers   = (const float*)d_in[2];
  const float* distances = (const float*)d_in[3];
  const float* W_op      = (const float*)d_in[4];
  const float* b_op      = (const float*)d_in[5];
  const float* W_agg     = (const float*)d_in[6];
  const float* b_agg     = (const float*)d_in[7];
  float* out = (float*)d_out;
  int* ids = (int*)d_ws;  // B*M*K int32 = 512 KB scratch

  const int rows = Bdim * Mctr;  // 4096 centers total
  ball_query_kernel<<<rows / 8, 256, 0, stream>>>(distances, ids);
  pointnet_fused_kernel<<<rows / WAVES2, 32 * WAVES2, 0, stream>>>(
      positions, features, centers, ids, W_op, b_op, W_agg, b_agg, out);
}